// SphericalFourierLayer_9818295239409
// MI455X (gfx1250) — compile-verified
//
#include <hip/hip_runtime.h>

typedef float v2f __attribute__((ext_vector_type(2)));
typedef float v8f __attribute__((ext_vector_type(8)));

#define SNLAT 128
#define SNLON 256
#define SCIN  128
#define SCOUT 128
#define SMODES 32
#define SBATCH 4

// ---------------------------------------------------------------------------
// Prep: DFT analysis table Bdft[n][j] (j<32: re, j>=32: im) and irfft
// synthesis table Birfft[j][n] with c_m and sign folded in.
// ---------------------------------------------------------------------------
__global__ void build_tables(float* __restrict__ Bdft, float* __restrict__ Birfft) {
    int idx = blockIdx.x * blockDim.x + threadIdx.x;
    const float TWO_PI = 6.2831853071795864769f;
    const float s = TWO_PI / (float)SNLON;
    if (idx < SNLON * 64) {
        // Bdft[n][j]: rfft * (2*pi/nlon): re = cos, im = -sin
        int n = idx >> 6, j = idx & 63;
        int m = j & 31;
        float th = s * (float)(m * n);
        Bdft[idx] = (j < 32) ? cosf(th) * s : -sinf(th) * s;
    } else {
        // Birfft[j][n]: out[n] = sum_m c_m*(xr*cos(th) - xi*sin(th))
        int k = idx - SNLON * 64;
        int j = k >> 8, n = k & 255;
        int m = j & 31;
        float c = (m == 0) ? 1.0f : 2.0f;
        float th = s * (float)(m * n);
        Birfft[k] = (j < 32) ? c * cosf(th) : -c * sinf(th);
    }
}

// Pw[m][t][l] = pct[l][m][t]*w[t]   (analysis operator, K=t rows, N=l cols)
// Psyn[m][l][t] = pct[l][m][t]      (synthesis operator, K=l rows, N=t cols)
__global__ void build_legendre(const float* __restrict__ pct, const float* __restrict__ w,
                               float* __restrict__ Pw, float* __restrict__ Psyn) {
    int idx = blockIdx.x * blockDim.x + threadIdx.x;   // 32*128*32
    int m = idx >> 12;
    int rem = idx & 4095;
    int t = rem >> 5;
    int l = rem & 31;
    // pct shape (128, 129, 128)
    float p = pct[(l * 129 + m) * 128 + t];
    Pw[idx] = p * w[t];                         // [m][t][l] == idx layout
    Psyn[(m * 32 + l) * 128 + t] = p;
}

// ---------------------------------------------------------------------------
// Generic fp32 WMMA GEMM:  C[z] = A[z] * B[z]
//   A strided (ar per row, ac per col, az per z; rows >= part_rows add a_extra)
//   B row-major [K][N] contiguous per z (bz per z)
//   C strided (cr, cc, cz; rows >= part_rows add c_extra)
// Block: 128 threads = 4 waves; each wave owns a 16x64 C strip; 64x64 C tile.
// Uses V_WMMA_F32_16X16X4_F32; fragment layouts per ISA 7.12.2:
//   A 16x4: VGPR v holds K=v (lanes 0-15) / K=v+2 (lanes 16-31)
//   B 4x16: VGPR v holds K=v (lanes 0-15) / K=v+2 (lanes 16-31), N = lane&15
//   C 16x16: VGPR r holds M=r (lanes 0-15) / M=r+8 (lanes 16-31)
// B is staged into LDS pre-swizzled in fragment order so each lane's
// (b.x, b.y) pair is one 8B-aligned ds_load_b64; A rows are padded to an
// even stride so the (a.x, a.y) pair is likewise a single b64 load.
// ---------------------------------------------------------------------------
__global__ __launch_bounds__(128)
void wmma_gemm_f32(const float* __restrict__ A, const float* __restrict__ B,
                   float* __restrict__ C,
                   int M, int N, int K,
                   long ar, long ac, long az, int part_rows, long a_extra,
                   long bz,
                   long cr, long cc, long cz, long c_extra) {
    __shared__ float As[64][18];        // row stride even -> 8B-aligned pairs
    __shared__ float Bsw[8][64][2];     // [kk*2+h][n][pair], fragment order

    const int tid  = threadIdx.x;
    const int lane = tid & 31;
    const int wave = tid >> 5;
    const int z    = blockIdx.z;
    const int row0 = blockIdx.y * 64;
    const int n0   = blockIdx.x * 64;

    const float* Az = A + az * (long)z;
    const float* Bz = B + bz * (long)z;
    float*       Cz = C + cz * (long)z;

    v8f acc[4];
    #pragma unroll
    for (int ct = 0; ct < 4; ++ct)
        #pragma unroll
        for (int e = 0; e < 8; ++e) acc[ct][e] = 0.0f;

    const int nn = lane & 15;
    const int h  = (lane >= 16) ? 1 : 0;   // K-half select (per ISA layout)

    for (int k0 = 0; k0 < K; k0 += 16) {
        // stage A tile 64x16 (row-major, even-padded rows)
        for (int i = tid; i < 64 * 16; i += 128) {
            int r = i >> 4, c = i & 15;
            int gr = row0 + r;
            long off = (long)gr * ar + (long)(k0 + c) * ac;
            if (part_rows && gr >= part_rows) off += a_extra;
            As[r][c] = Az[off];
        }
        // stage B tile 16x64, swizzled: row r = kk*4 + 2h + pos
        for (int i = tid; i < 16 * 64; i += 128) {
            int r = i >> 6, c = i & 63;
            int gc = n0 + c;
            float v = (gc < N) ? Bz[(long)(k0 + r) * (long)N + gc] : 0.0f;
            int kk  = r >> 2;
            int hh  = (r >> 1) & 1;
            int pos = r & 1;
            Bsw[kk * 2 + hh][c][pos] = v;
        }
        __syncthreads();

        #pragma unroll
        for (int kk = 0; kk < 4; ++kk) {
            v2f a = *(const v2f*)&As[wave * 16 + nn][kk * 4 + 2 * h];
            #pragma unroll
            for (int ct = 0; ct < 4; ++ct) {
                v2f b = *(const v2f*)&Bsw[kk * 2 + h][ct * 16 + nn][0];
                acc[ct] = __builtin_amdgcn_wmma_f32_16x16x4_f32(
                    false, a, false, b, (short)0, acc[ct], false, false);
            }
        }
        __syncthreads();
    }

    // store C
    const int rhalf = (lane >= 16) ? 8 : 0;
    #pragma unroll
    for (int ct = 0; ct < 4; ++ct) {
        int gc = n0 + ct * 16 + (lane & 15);
        if (gc >= N) continue;
        #pragma unroll
        for (int r = 0; r < 8; ++r) {
            int gr = row0 + wave * 16 + r + rhalf;
            long off = (long)gr * cr + (long)gc * cc;
            if (part_rows && gr >= part_rows) off += c_extra;
            Cz[off] = acc[ct][r];
        }
    }
}

// ---------------------------------------------------------------------------
// Spectral conv: out[b,o,l,m] = sum_i x_spec[b,i,l,m] * (wr + i*wi)
// M=4 batched GEMV -> weight-bandwidth bound; thread map puts contiguous
// (m,part) weight dimension across the wave for fully coalesced 256B reads.
// spec layout:    [m][l][part][b*128 + i]     (from analysis GEMM)
// specOut layout: [m][part][b*128 + o][l]     (feeds synthesis GEMM)
// ---------------------------------------------------------------------------
__global__ __launch_bounds__(256)
void spectral_conv(const float* __restrict__ spec, const float* __restrict__ Wgt,
                   float* __restrict__ specOut) {
    const int tid = threadIdx.x;
    const int m = tid & 31;                    // contiguous in weight memory
    const int o = (blockIdx.x << 3) + (tid >> 5);
    const int l = blockIdx.y;

    const float* xr_base = spec + ((long)(m * 32 + l) * 2 + 0) * 512;
    const float* xi_base = xr_base + 512;

    float orr[4] = {0.f, 0.f, 0.f, 0.f};
    float oii[4] = {0.f, 0.f, 0.f, 0.f};

    const long istride = (long)SCOUT * SMODES * SMODES * 2;  // per-i weight stride
    long widx = ((((long)o) * SMODES + l) * SMODES + m) * 2;

    for (int i = 0; i < SCIN; ++i, widx += istride) {
        __builtin_prefetch(Wgt + widx + 8 * istride, 0, 1);  // global_prefetch_b8
        const float2 wv = *(const float2*)(Wgt + widx);      // (wr, wi) 8B aligned
        #pragma unroll
        for (int b = 0; b < 4; ++b) {
            float xr = xr_base[b * 128 + i];
            float xi = xi_base[b * 128 + i];
            orr[b] = fmaf(xr, wv.x, fmaf(-xi, wv.y, orr[b]));
            oii[b] = fmaf(xr, wv.y, fmaf( xi, wv.x, oii[b]));
        }
    }
    #pragma unroll
    for (int b = 0; b < 4; ++b) {
        specOut[((long)(m * 2 + 0) * 512 + b * 128 + o) * 32 + l] = orr[b];
        specOut[((long)(m * 2 + 1) * 512 + b * 128 + o) * 32 + l] = oii[b];
    }
}

// ---------------------------------------------------------------------------
extern "C" void kernel_launch(void* const* d_in, const int* in_sizes, int n_in,
                              void* d_out, int out_size, void* d_ws, size_t ws_size,
                              hipStream_t stream) {
    const float* x   = (const float*)d_in[0];   // (4,128,128,256)
    const float* wgt = (const float*)d_in[1];   // (128,128,32,32,2)
    const float* pct = (const float*)d_in[2];   // (128,129,128)
    const float* w   = (const float*)d_in[3];   // (128,)
    float* out = (float*)d_out;                 // (4,128,128,256)
    float* ws  = (float*)d_ws;

    float* Bdft    = ws;                    // 256*64      = 16384
    float* Birfft  = Bdft    + 16384;       // 64*256      = 16384
    float* Pw      = Birfft  + 16384;       // 32*128*32   = 131072
    float* Psyn    = Pw      + 131072;      // 32*32*128   = 131072
    float* xfT     = Psyn    + 131072;      // 64*512*128  = 4194304
    float* spec    = xfT     + 4194304;     // 32*32*2*512 = 1048576
    float* specOut = spec    + 1048576;     // 32*2*512*32 = 1048576
    float* xfoT    = specOut + 1048576;     // 64*512*128  = 4194304
    // total ~ 10.78M floats ~= 43.1 MB of workspace

    build_tables  <<<(2 * 16384) / 256, 256, 0, stream>>>(Bdft, Birfft);
    build_legendre<<<131072 / 256,      256, 0, stream>>>(pct, w, Pw, Psyn);

    // Stage 1: DFT analysis  xfT[j][bct] = x[bct][n] * Bdft[n][j]
    wmma_gemm_f32<<<dim3(1, 1024, 1), 128, 0, stream>>>(
        x, Bdft, xfT,
        65536, 64, 256,
        /*ar*/256, /*ac*/1, /*az*/0, /*part_rows*/0, /*a_extra*/0,
        /*bz*/0,
        /*cr*/1, /*cc*/65536, /*cz*/0, /*c_extra*/0);

    // Stage 2: Legendre analysis per m: spec[m][l][part][bc]
    //   A rows: 0..511 = xfT[m] (re), 512..1023 = xfT[m+32] (im)
    wmma_gemm_f32<<<dim3(1, 16, 32), 128, 0, stream>>>(
        xfT, Pw, spec,
        1024, 32, 128,
        /*ar*/128, /*ac*/1, /*az*/65536, /*part_rows*/512, /*a_extra*/2031616L,
        /*bz*/4096,
        /*cr*/1, /*cc*/1024, /*cz*/32768, /*c_extra*/0);

    // Stage 3: complex spectral conv (weight-BW bound, VALU)
    spectral_conv<<<dim3(16, 32, 1), 256, 0, stream>>>(spec, wgt, specOut);

    // Stage 4: Legendre synthesis per m: xfoT[m(+32*part)][bo][t]
    wmma_gemm_f32<<<dim3(2, 16, 32), 128, 0, stream>>>(
        specOut, Psyn, xfoT,
        1024, 128, 32,
        /*ar*/32, /*ac*/1, /*az*/32768, /*part_rows*/512, /*a_extra*/0,
        /*bz*/4096,
        /*cr*/128, /*cc*/1, /*cz*/65536, /*c_extra*/2031616L);

    // Stage 5: irfft synthesis  out[bot][n] = xfoT[j][bot] * Birfft[j][n]
    wmma_gemm_f32<<<dim3(4, 1024, 1), 128, 0, stream>>>(
        xfoT, Birfft, out,
        65536, 256, 64,
        /*ar*/1, /*ac*/65536, /*az*/0, /*part_rows*/0, /*a_extra*/0,
        /*bz*/0,
        /*cr*/256, /*cc*/1, /*cz*/0, /*c_extra*/0);
}